// LocalGraph_37692632989994
// MI455X (gfx1250) — compile-verified
//
#include <hip/hip_runtime.h>
#include <stdint.h>

typedef __attribute__((ext_vector_type(2))) float v2f;
typedef __attribute__((ext_vector_type(8))) float v8f;

// ---------------- JAX threefry2x32 (exact) ----------------
__device__ __forceinline__ unsigned rotl32(unsigned x, unsigned r) {
  return (x << r) | (x >> (32u - r));
}

__device__ __forceinline__ void threefry2x32(unsigned k0, unsigned k1,
                                             unsigned& x0, unsigned& x1) {
  unsigned ks2 = k0 ^ k1 ^ 0x1BD11BDAu;
  x0 += k0; x1 += k1;
#define TF_R(rr) { x0 += x1; x1 = rotl32(x1, rr); x1 ^= x0; }
  TF_R(13u) TF_R(15u) TF_R(26u) TF_R(6u)
  x0 += k1;  x1 += ks2 + 1u;
  TF_R(17u) TF_R(29u) TF_R(16u) TF_R(24u)
  x0 += ks2; x1 += k0 + 2u;
  TF_R(13u) TF_R(15u) TF_R(26u) TF_R(6u)
  x0 += k0;  x1 += k1 + 3u;
  TF_R(17u) TF_R(29u) TF_R(16u) TF_R(24u)
  x0 += k1;  x1 += ks2 + 4u;
  TF_R(13u) TF_R(15u) TF_R(26u) TF_R(6u)
  x0 += ks2; x1 += k0 + 5u;
#undef TF_R
}

__device__ __forceinline__ float u01(unsigned b) {
  return __uint_as_float((b >> 9) | 0x3f800000u) - 1.0f;
}

// ---------------- utility ----------------
__global__ void zeroK(float* p, int n) {
  int i = blockIdx.x * blockDim.x + threadIdx.x;
  if (i < n) p[i] = 0.0f;
}

// ---------------- edge masks + order sums ----------------
// mask bit d = cumulative keep through dropout level d (d=0..2)
__global__ __launch_bounds__(256) void maskOrderK(const int* rows, const float* adj,
                                                  unsigned char* mask,
                                                  float* ord0, float* ord1, float* ord2,
                                                  int E2) {
  int t = blockIdx.x * blockDim.x + threadIdx.x;
  if (t >= E2) return;
  int eLo = t, eHi = t + E2;
  unsigned mLo = 0u, mHi = 0u;
  bool cLo = true, cHi = true;
  const float P[3] = {0.5f, 0.25f, 0.125f};
#pragma unroll
  for (int d = 0; d < 3; ++d) {
    // fold_in(key(42), d) : threefry((0,42), (0,d))
    unsigned K0 = 0u, K1 = (unsigned)d;
    threefry2x32(0u, 42u, K0, K1);
    // uniform(E): counts split in halves
    unsigned b0 = (unsigned)eLo, b1 = (unsigned)eHi;
    threefry2x32(K0, K1, b0, b1);
    if (!(u01(b0) + P[d] >= 1.0f)) cLo = false;   // keep = floor(u+p)
    if (!(u01(b1) + P[d] >= 1.0f)) cHi = false;
    if (cLo) mLo |= (1u << d);
    if (cHi) mHi |= (1u << d);
  }
  mask[eLo] = (unsigned char)mLo;
  mask[eHi] = (unsigned char)mHi;

  int rLo = rows[eLo], rHi = rows[eHi];
  float vLo = adj[eLo], vHi = adj[eHi];
  atomicAdd(&ord0[rLo], vLo);              atomicAdd(&ord0[rHi], vHi);
  if (mLo & 1u) atomicAdd(&ord1[rLo], vLo);
  if (mHi & 1u) atomicAdd(&ord1[rHi], vHi);
  if (mLo & 2u) atomicAdd(&ord2[rLo], vLo);
  if (mHi & 2u) atomicAdd(&ord2[rHi], vHi);
}

// ---------------- SpMM scatter: one wave32 per edge, lane = dim ----------------
__global__ __launch_bounds__(256) void scatterK(const int* rows, const int* cols,
                                                const float* adj, const unsigned char* mask,
                                                int bit, const float* src, const float* numSrc,
                                                float* Bacc, float* numB, int E) {
  int w = blockIdx.x * (blockDim.x >> 5) + (threadIdx.x >> 5);
  int lane = threadIdx.x & 31;
  if (w >= E) return;
  if (bit >= 0 && !((mask[w] >> bit) & 1)) return;
  int r = rows[w], c = cols[w];
  float v = adj[w];
  atomicAdd(&Bacc[(size_t)r * 32 + lane], v * src[(size_t)c * 32 + lane]);
  if (numSrc != nullptr && lane == 0) atomicAdd(&numB[r], v * numSrc[c]);
}

// ---------------- pointwise: level 0 ----------------
__global__ void point0K(const float* X, float* Bacc, float* embP, float* SUMe,
                        const float* ord0, float* SUMn, float* numP, float* numB, int n32) {
  int i = blockIdx.x * blockDim.x + threadIdx.x;
  if (i >= n32) return;
  int n = i >> 5;
  float v = Bacc[i] - X[i];      // fst_emb = spmm(embeds) - embeds
  embP[i] = v; SUMe[i] = v; Bacc[i] = 0.0f;
  if ((i & 31) == 0) {
    float o = ord0[n];
    SUMn[n] = o; numP[n] = o; numB[n] = 0.0f;   // fst_num = order
  }
}

// ---------------- pointwise: levels 1..3 ----------------
__global__ void pointK(float* Bacc, float* embP, float* SUMe, const float* ord,
                       float* SUMn, float* numP, float* numB, int n32) {
  int i = blockIdx.x * blockDim.x + threadIdx.x;
  if (i >= n32) return;
  int n = i >> 5;
  float ep = embP[i], o = ord[n];
  float v = Bacc[i] - ep - o * ep;
  embP[i] = v; SUMe[i] += v; Bacc[i] = 0.0f;
  if ((i & 31) == 0) {
    float np = numP[n];
    float nv = numB[n] - np - o;
    SUMn[n] += nv; numP[n] = nv; numB[n] = 0.0f;
  }
}

// ---------------- scores via V_WMMA_F32_16X16X4_F32 chains + Gumbel ----------------
// Wave handles 16 nodes. A(16x4 f32) lane layout == B(4x16 f32) lane layout for the
// same node-index base, so one v2f register pair serves as both operands.
__global__ __launch_bounds__(256) void scoreK(const float* SUMe, const float* SUMn,
                                              const float* X, float* scores, int N) {
  int wave = blockIdx.x * (blockDim.x >> 5) + (threadIdx.x >> 5);
  int lane = threadIdx.x & 31;
  int base = wave * 16;
  if (base >= N) return;                 // uniform per wave; EXEC stays all-ones
  int r = lane & 15, half = lane >> 4;
  int node = base + r;
  float w = 1.0f / (SUMn[node] + 1e-8f);
  const float* Sp = SUMe + (size_t)node * 32;
  const float* Xp = X + (size_t)node * 32;
  v8f cgx = {}; v8f cgg = {}; v8f cxx = {};
#pragma unroll
  for (int s = 0; s < 8; ++s) {
    int d0 = 4 * s + 2 * half;
    v2f a = { Sp[d0] * w, Sp[d0 + 1] * w };   // G = S / (num+eps)
    v2f b = { Xp[d0],     Xp[d0 + 1] };
    cgx = __builtin_amdgcn_wmma_f32_16x16x4_f32(false, a, false, b, (short)0, cgx, false, false);
    cgg = __builtin_amdgcn_wmma_f32_16x16x4_f32(false, a, false, a, (short)0, cgg, false, false);
    cxx = __builtin_amdgcn_wmma_f32_16x16x4_f32(false, b, false, b, (short)0, cxx, false, false);
  }
  // Diagonal of 16x16 C: m<8 -> (vgpr m, lane m); m>=8 -> (vgpr m-8, lane m+16)
  bool act = (lane < 8) || (lane >= 24);
  int j = (lane < 8) ? lane : lane - 24;
  int m = (lane < 8) ? lane : lane - 16;
  float dgx = 0.0f, dgg = 0.0f, dxx = 0.0f;
#pragma unroll
  for (int k = 0; k < 8; ++k) {
    if (j == k) { dgx = cgx[k]; dgg = cgg[k]; dxx = cxx[k]; }
  }
  if (act) {
    float sc = dgx / (fmaxf(sqrtf(dgg), 1e-12f) * fmaxf(sqrtf(dxx), 1e-12f));
    int nm = base + m;
    int N2 = N >> 1;
    unsigned x0 = (nm < N2) ? (unsigned)nm : (unsigned)(nm - N2);
    unsigned x1 = (nm < N2) ? (unsigned)(nm + N2) : (unsigned)nm;
    threefry2x32(0u, 7u, x0, x1);          // key(7) uniform(N)
    float u = u01((nm < N2) ? x0 : x1);
    sc += -logf(-logf(u));                 // Gumbel
    scores[nm] = sc;
  }
}

// ---------------- exact top-K via 4-round byte radix select ----------------
// sel[0]=prefix  sel[1]=kRem  sel[2..257]=hist  sel[258]=cntGreater sel[259]=cntEqual
struct UPair { unsigned k; unsigned i; };

__device__ __forceinline__ unsigned fkey(float f) {
  unsigned u = __float_as_uint(f);
  return (u & 0x80000000u) ? ~u : (u | 0x80000000u);
}

__global__ void selInitK(unsigned* sel, int K) {
  int t = threadIdx.x;
  if (t == 0) { sel[0] = 0u; sel[1] = (unsigned)K; sel[258] = 0u; sel[259] = 0u; }
  sel[2 + t] = 0u;
}

__global__ __launch_bounds__(256) void histK(const float* scores, unsigned* sel, int rnd, int N) {
  __shared__ unsigned h[256];
  h[threadIdx.x] = 0u;
  __syncthreads();
  int i = blockIdx.x * blockDim.x + threadIdx.x;
  if (i < N) {
    unsigned key = fkey(scores[i]);
    bool ok = (rnd == 0) || ((key >> (32 - 8 * rnd)) == sel[0]);
    if (ok) atomicAdd(&h[(key >> (24 - 8 * rnd)) & 255u], 1u);
  }
  __syncthreads();
  unsigned c = h[threadIdx.x];
  if (c) atomicAdd(&sel[2 + threadIdx.x], c);
}

__global__ void scanK(unsigned* sel) {
  if (threadIdx.x == 0) {
    unsigned kRem = sel[1], prefix = sel[0], cum = 0u;
    for (int b = 255; b >= 0; --b) {
      unsigned c = sel[2 + b];
      if (cum + c >= kRem) { sel[0] = (prefix << 8) | (unsigned)b; sel[1] = kRem - cum; break; }
      cum += c;
    }
  }
  __syncthreads();
  sel[2 + threadIdx.x] = 0u;
}

__global__ __launch_bounds__(256) void collectK(const float* scores, unsigned* sel,
                                                UPair* pairs, int N, int K) {
  int i = blockIdx.x * blockDim.x + threadIdx.x;
  if (i >= N) return;
  unsigned key = fkey(scores[i]);
  unsigned T = sel[0], kRem = sel[1];
  if (key > T) {
    unsigned p = atomicAdd(&sel[258], 1u);
    if (p < (unsigned)K) { pairs[p].k = key; pairs[p].i = (unsigned)i; }
  } else if (key == T) {
    unsigned p = atomicAdd(&sel[259], 1u);
    if (p < kRem) { unsigned s = (unsigned)K - kRem + p; pairs[s].k = key; pairs[s].i = (unsigned)i; }
  }
}

// 1024-element bitonic sort in LDS: descending by key, ties ascending index
__global__ __launch_bounds__(1024) void sortK(const UPair* pairs, int* cand) {
  __shared__ unsigned long long s[1024];
  int t = threadIdx.x;
  UPair p = pairs[t];
  s[t] = ((unsigned long long)p.k << 32) | (unsigned long long)(unsigned)(~p.i);
  __syncthreads();
  for (int k = 2; k <= 1024; k <<= 1) {
    for (int j = k >> 1; j > 0; j >>= 1) {
      int x = t ^ j;
      if (x > t) {
        unsigned long long A = s[t], B = s[x];
        bool desc = ((t & k) == 0);
        if (desc ? (A < B) : (A > B)) { s[t] = B; s[x] = A; }
      }
      __syncthreads();
    }
  }
  cand[t] = (int)(~(unsigned)(s[t] & 0xFFFFFFFFull));
}

// ---------------- host-side launch sequence (graph-capture safe) ----------------
extern "C" void kernel_launch(void* const* d_in, const int* in_sizes, int n_in,
                              void* d_out, int out_size, void* d_ws, size_t ws_size,
                              hipStream_t stream) {
  const int* edge_index = (const int*)d_in[0];
  const float* adj = (const float*)d_in[1];
  const float* X = (const float*)d_in[2];
  const int E = in_sizes[0] / 2;
  const int N = in_sizes[2] / 32;
  const int K = out_size - N;          // 1024
  const int* rows = edge_index;
  const int* cols = edge_index + E;

  float* ws   = (float*)d_ws;
  float* ord0 = ws;                    // N   (contiguous zero region start)
  float* ord1 = ord0 + N;              // N
  float* ord2 = ord1 + N;              // N
  float* numB = ord2 + N;              // N
  float* Bacc = numB + N;              // 32N (zero region end: 36N floats)
  float* embP = Bacc + (size_t)32 * N; // 32N
  float* SUMe = embP + (size_t)32 * N; // 32N
  float* SUMn = SUMe + (size_t)32 * N; // N
  float* numP = SUMn + N;              // N
  unsigned* sel = (unsigned*)(numP + N);       // 260 u32
  UPair* pairs  = (UPair*)(sel + 260);         // K pairs
  unsigned char* mask = (unsigned char*)(pairs + K); // E bytes

  float* scores = (float*)d_out;
  int* cand = (int*)d_out + N;

  const int TPB = 256;
  int zeroN = 36 * N;
  zeroK<<<(zeroN + TPB - 1) / TPB, TPB, 0, stream>>>(ord0, zeroN);

  int E2 = E / 2;
  maskOrderK<<<(E2 + TPB - 1) / TPB, TPB, 0, stream>>>(rows, adj, mask, ord0, ord1, ord2, E2);

  int edgeBlocks = (E + 7) / 8;        // 8 waves per 256-thread block, 1 wave per edge
  int n32 = 32 * N;
  int pwBlocks = (n32 + TPB - 1) / TPB;

  // level 0
  scatterK<<<edgeBlocks, TPB, 0, stream>>>(rows, cols, adj, mask, -1, X, nullptr, Bacc, numB, E);
  point0K<<<pwBlocks, TPB, 0, stream>>>(X, Bacc, embP, SUMe, ord0, SUMn, numP, numB, n32);
  // level 1 (mask bit 0, order0)
  scatterK<<<edgeBlocks, TPB, 0, stream>>>(rows, cols, adj, mask, 0, embP, numP, Bacc, numB, E);
  pointK<<<pwBlocks, TPB, 0, stream>>>(Bacc, embP, SUMe, ord0, SUMn, numP, numB, n32);
  // level 2 (mask bit 1, order1)
  scatterK<<<edgeBlocks, TPB, 0, stream>>>(rows, cols, adj, mask, 1, embP, numP, Bacc, numB, E);
  pointK<<<pwBlocks, TPB, 0, stream>>>(Bacc, embP, SUMe, ord1, SUMn, numP, numB, n32);
  // level 3 (mask bit 2, order2)
  scatterK<<<edgeBlocks, TPB, 0, stream>>>(rows, cols, adj, mask, 2, embP, numP, Bacc, numB, E);
  pointK<<<pwBlocks, TPB, 0, stream>>>(Bacc, embP, SUMe, ord2, SUMn, numP, numB, n32);

  // scores (WMMA) + Gumbel
  int tiles = (N + 15) / 16;
  scoreK<<<(tiles + 7) / 8, TPB, 0, stream>>>(SUMe, SUMn, X, scores, N);

  // exact top-K
  selInitK<<<1, 256, 0, stream>>>(sel, K);
  int hBlocks = (N + TPB - 1) / TPB;
  for (int r = 0; r < 4; ++r) {
    histK<<<hBlocks, TPB, 0, stream>>>(scores, sel, r, N);
    scanK<<<1, 256, 0, stream>>>(sel);
  }
  collectK<<<hBlocks, TPB, 0, stream>>>(scores, sel, pairs, N, K);
  sortK<<<1, 1024, 0, stream>>>(pairs, cand);
}